// GNN_5952824672568
// MI455X (gfx1250) — compile-verified
//
#include <hip/hip_runtime.h>
#include <math.h>

typedef __attribute__((ext_vector_type(2))) float v2f;
typedef __attribute__((ext_vector_type(8))) float v8f;

#define HEADS   4
#define CHAN    128
#define DDIM    512
#define OUTD    256
#define NEG_ATT 0.2f
#define NEG_ACT 0.01f

static inline int cdiv64(long long a, long long b) { return (int)((a + b - 1) / b); }

// ---------------------------------------------------------------------------
// f32 WMMA GEMM: C[M,N] = A[M,K] @ B[K,N] (+ optional per-column bias)
// 256 threads = 8 waves; block tile 32(M) x 64(N); BK = 32; wave tile 16x16.
// Requires M%32==0, N%64==0, K%32==0 (true here: M=20000, N∈{512,256}, K∈{128,512}).
// ---------------------------------------------------------------------------
__global__ __launch_bounds__(256) void k_gemm_f32(
    const float* __restrict__ A, const float* __restrict__ B,
    float* __restrict__ Cm, const float* __restrict__ bias,
    int M, int N, int K)
{
    __shared__ float As[32][33];   // +1 pad: avoid bank conflicts on column reads
    __shared__ float Bs[32][65];

    const int tid  = threadIdx.x;
    const int lane = tid & 31;
    const int wave = tid >> 5;
    const int wm   = wave >> 2;            // 0..1  (M sub-tile)
    const int wn   = wave & 3;             // 0..3  (N sub-tile)
    const int bm   = blockIdx.y * 32;
    const int bn   = blockIdx.x * 64;

    v8f acc = {};

    const int mrow  = wm * 16 + (lane & 15);
    const int ncol  = wn * 16 + (lane & 15);
    const int khalf = (lane >> 4) << 1;    // lanes 0-15 -> K+0/1, lanes 16-31 -> K+2/3

    for (int k0 = 0; k0 < K; k0 += 32) {
        for (int i = tid; i < 32 * 32; i += 256) {
            int r = i >> 5, c = i & 31;
            As[r][c] = A[(size_t)(bm + r) * K + (k0 + c)];
        }
        for (int i = tid; i < 32 * 64; i += 256) {
            int r = i >> 6, c = i & 63;
            Bs[r][c] = B[(size_t)(k0 + r) * N + (bn + c)];
        }
        __syncthreads();
#pragma unroll
        for (int kk = 0; kk < 32; kk += 4) {
            v2f a, b;
            a.x = As[mrow][kk + khalf + 0];
            a.y = As[mrow][kk + khalf + 1];
            b.x = Bs[kk + khalf + 0][ncol];
            b.y = Bs[kk + khalf + 1][ncol];
            acc = __builtin_amdgcn_wmma_f32_16x16x4_f32(
                      false, a, false, b, (short)0, acc, false, false);
        }
        __syncthreads();
    }

    const int col  = bn + wn * 16 + (lane & 15);
    const int row0 = bm + wm * 16 + ((lane >> 4) << 3);   // lanes 16-31 hold M+8..M+15
    const float bv = bias ? bias[col] : 0.0f;
#pragma unroll
    for (int v = 0; v < 8; ++v)
        Cm[(size_t)(row0 + v) * N + col] = acc[v] + bv;
}

// ---------------------------------------------------------------------------
// Elementwise / graph kernels
// ---------------------------------------------------------------------------
__global__ void k_concat(const float* __restrict__ X, const float* __restrict__ pos,
                         float* __restrict__ xcat, int N)
{
    int i = blockIdx.x * blockDim.x + threadIdx.x;
    if (i >= N * 128) return;
    int n = i >> 7, c = i & 127;
    xcat[i] = (c < 127) ? X[(size_t)n * 127 + c] : pos[n];
}

__global__ void k_fill(float* __restrict__ p, float v, int n)
{
    int i = blockIdx.x * blockDim.x + threadIdx.x;
    if (i < n) p[i] = v;
}

// al_s[n,h] = sum_c xl[n,h,c]*a_src[h,c] ; al_d likewise
__global__ void k_att_scores(const float* __restrict__ xl,
                             const float* __restrict__ a_src, const float* __restrict__ a_dst,
                             float* __restrict__ als, float* __restrict__ ald, int N)
{
    int t = blockIdx.x * blockDim.x + threadIdx.x;
    if (t >= N * HEADS) return;
    int n = t >> 2, h = t & 3;
    const float* row = xl + (size_t)n * DDIM + h * CHAN;
    const float* as  = a_src + h * CHAN;
    const float* ad  = a_dst + h * CHAN;
    float s = 0.f, d = 0.f;
#pragma unroll 4
    for (int c = 0; c < CHAN; ++c) { float v = row[c]; s += v * as[c]; d += v * ad[c]; }
    als[t] = s; ald[t] = d;
}

__device__ inline void atomic_max_f32(float* addr, float v)
{
    // IEEE-ordered max via integer atomics (works for mixed signs with -1e30 init)
    if (v >= 0.0f) atomicMax((int*)addr, __float_as_int(v));
    else           atomicMin((unsigned int*)addr, (unsigned int)__float_as_int(v));
}

__device__ inline void edge_endpoints(int e, const int* __restrict__ src,
                                      const int* __restrict__ dst, int E, int& s, int& d)
{
    if (e < E) { s = src[e]; d = dst[e]; }
    else       { s = e - E; d = s; }       // appended self-loops
}

__global__ void k_edge_max(const int* __restrict__ src, const int* __restrict__ dst,
                           const float* __restrict__ als, const float* __restrict__ ald,
                           float* __restrict__ mseg, int E, int N)
{
    int e = blockIdx.x * blockDim.x + threadIdx.x;
    if (e >= E + N) return;
    int s, d; edge_endpoints(e, src, dst, E, s, d);
#pragma unroll
    for (int h = 0; h < HEADS; ++h) {
        float l = als[s * HEADS + h] + ald[d * HEADS + h];
        l = (l > 0.f) ? l : NEG_ATT * l;
        atomic_max_f32(&mseg[d * HEADS + h], l);
    }
}

__global__ void k_edge_exp(const int* __restrict__ src, const int* __restrict__ dst,
                           const float* __restrict__ als, const float* __restrict__ ald,
                           const float* __restrict__ mseg, float* __restrict__ sseg,
                           float* __restrict__ ealpha, int E, int N)
{
    int e = blockIdx.x * blockDim.x + threadIdx.x;
    if (e >= E + N) return;
    int s, d; edge_endpoints(e, src, dst, E, s, d);
#pragma unroll
    for (int h = 0; h < HEADS; ++h) {
        float l = als[s * HEADS + h] + ald[d * HEADS + h];
        l = (l > 0.f) ? l : NEG_ATT * l;
        float ev = expf(l - mseg[d * HEADS + h]);
        ealpha[(size_t)e * HEADS + h] = ev;
        atomicAdd(&sseg[d * HEADS + h], ev);
    }
}

// one thread per (edge, feature): agg[dst, f] += xl[src, f] * alpha[e, f>>7]
__global__ void k_edge_scatter(const int* __restrict__ src, const int* __restrict__ dst,
                               const float* __restrict__ xl, const float* __restrict__ ealpha,
                               const float* __restrict__ sseg, float* __restrict__ agg,
                               int E, int N)
{
    long long w = (long long)blockIdx.x * blockDim.x + threadIdx.x;
    long long total = (long long)(E + N) * DDIM;
    if (w >= total) return;
    int e = (int)(w >> 9);
    int f = (int)(w & (DDIM - 1));
    int h = f >> 7;
    int s, d; edge_endpoints(e, src, dst, E, s, d);
    float alpha = ealpha[(size_t)e * HEADS + h] / (sseg[d * HEADS + h] + 1e-16f);
    atomicAdd(&agg[(size_t)d * DDIM + f], xl[(size_t)s * DDIM + f] * alpha);
}

// per-channel sum / sum-of-squares; block handles strided rows, thread owns 2 channels
__global__ __launch_bounds__(256) void k_bn_stats(const float* __restrict__ h,
                                                  float* __restrict__ bnsum,
                                                  float* __restrict__ bnsq, int N)
{
    int c0 = threadIdx.x, c1 = threadIdx.x + 256;
    float s0 = 0.f, q0 = 0.f, s1 = 0.f, q1 = 0.f;
    for (int r = blockIdx.x; r < N; r += gridDim.x) {
        float v0 = h[(size_t)r * DDIM + c0];
        float v1 = h[(size_t)r * DDIM + c1];
        s0 += v0; q0 += v0 * v0;
        s1 += v1; q1 += v1 * v1;
    }
    atomicAdd(&bnsum[c0], s0); atomicAdd(&bnsq[c0], q0);
    atomicAdd(&bnsum[c1], s1); atomicAdd(&bnsq[c1], q1);
}

// normalize + affine + LeakyReLU(0.01) + optional residual; in place
__global__ void k_bn_apply(float* __restrict__ h, const float* __restrict__ bnsum,
                           const float* __restrict__ bnsq, const float* __restrict__ g,
                           const float* __restrict__ b, const float* __restrict__ residual,
                           int N)
{
    long long i = (long long)blockIdx.x * blockDim.x + threadIdx.x;
    if (i >= (long long)N * DDIM) return;
    int c = (int)(i & (DDIM - 1));
    float invN  = 1.0f / (float)N;
    float mean  = bnsum[c] * invN;
    float var   = bnsq[c] * invN - mean * mean;
    float v = (h[i] - mean) * rsqrtf(var + 1e-5f) * g[c] + b[c];
    v = (v > 0.f) ? v : NEG_ACT * v;
    if (residual) v += residual[i];
    h[i] = v;
}

// ---------------------------------------------------------------------------
static void gat_aggregate(hipStream_t stream, const float* xl,
                          const int* src, const int* dst,
                          const float* a_s, const float* a_d,
                          int E, int N,
                          float* als, float* ald, float* mseg, float* sseg,
                          float* ealpha, float* agg)
{
    int Etot = E + N;
    k_att_scores<<<cdiv64((long long)N * HEADS, 256), 256, 0, stream>>>(xl, a_s, a_d, als, ald, N);
    k_fill<<<cdiv64((long long)N * HEADS, 256), 256, 0, stream>>>(mseg, -1e30f, N * HEADS);
    hipMemsetAsync(sseg, 0, (size_t)N * HEADS * sizeof(float), stream);
    k_edge_max<<<cdiv64(Etot, 256), 256, 0, stream>>>(src, dst, als, ald, mseg, E, N);
    k_edge_exp<<<cdiv64(Etot, 256), 256, 0, stream>>>(src, dst, als, ald, mseg, sseg, ealpha, E, N);
    hipMemsetAsync(agg, 0, (size_t)N * DDIM * sizeof(float), stream);
    k_edge_scatter<<<cdiv64((long long)Etot * DDIM, 256), 256, 0, stream>>>(
        src, dst, xl, ealpha, sseg, agg, E, N);
}

extern "C" void kernel_launch(void* const* d_in, const int* in_sizes, int n_in,
                              void* d_out, int out_size, void* d_ws, size_t ws_size,
                              hipStream_t stream)
{
    const float* X    = (const float*)d_in[0];
    const int*   ei   = (const int*)d_in[1];     // [2,E] (edge_weight d_in[2] unused)
    const float* pos  = (const float*)d_in[3];
    const float* W1   = (const float*)d_in[4];
    const float* as1  = (const float*)d_in[5];
    const float* ad1  = (const float*)d_in[6];
    const float* W2   = (const float*)d_in[8];
    const float* as2  = (const float*)d_in[9];
    const float* ad2  = (const float*)d_in[10];
    const float* bn1g = (const float*)d_in[12];
    const float* bn1b = (const float*)d_in[13];
    const float* bn2g = (const float*)d_in[14];
    const float* bn2b = (const float*)d_in[15];
    // d_in[16]/[17] (dm_W, dm_b) unused by the reference
    const float* outW = (const float*)d_in[18];
    const float* outb = (const float*)d_in[19];

    const int N = in_sizes[0] / 127;   // 20000
    const int E = in_sizes[1] / 2;     // 320000
    const int* src = ei;
    const int* dst = ei + E;
    const int Etot = E + N;

    // workspace layout (~140 MB of f32)
    float* p      = (float*)d_ws;
    float* x_cat  = p; p += (size_t)N * 128;
    float* xl     = p; p += (size_t)N * DDIM;
    float* hA     = p; p += (size_t)N * DDIM;    // layer-1 agg -> h / h_prev
    float* hB     = p; p += (size_t)N * DDIM;    // layer-2 agg -> final h
    float* als    = p; p += (size_t)N * HEADS;
    float* ald    = p; p += (size_t)N * HEADS;
    float* mseg   = p; p += (size_t)N * HEADS;
    float* sseg   = p; p += (size_t)N * HEADS;
    float* ealpha = p; p += (size_t)Etot * HEADS;
    float* bnsum  = p; p += DDIM;
    float* bnsq   = p; p += DDIM;
    (void)ws_size; (void)n_in; (void)out_size;

    // x = [X | pos_emb]
    k_concat<<<cdiv64((long long)N * 128, 256), 256, 0, stream>>>(X, pos, x_cat, N);

    // ---------------- layer 1 ----------------
    {
        dim3 grid(DDIM / 64, N / 32);
        k_gemm_f32<<<grid, 256, 0, stream>>>(x_cat, W1, xl, nullptr, N, DDIM, 128);
    }
    gat_aggregate(stream, xl, src, dst, as1, ad1, E, N, als, ald, mseg, sseg, ealpha, hA);
    hipMemsetAsync(bnsum, 0, DDIM * sizeof(float), stream);
    hipMemsetAsync(bnsq,  0, DDIM * sizeof(float), stream);
    k_bn_stats<<<512, 256, 0, stream>>>(hA, bnsum, bnsq, N);
    k_bn_apply<<<cdiv64((long long)N * DDIM, 256), 256, 0, stream>>>(
        hA, bnsum, bnsq, bn1g, bn1b, nullptr, N);          // hA now = h (= h_prev)

    // ---------------- layer 2 ----------------
    {
        dim3 grid(DDIM / 64, N / 32);
        k_gemm_f32<<<grid, 256, 0, stream>>>(hA, W2, xl, nullptr, N, DDIM, DDIM);
    }
    gat_aggregate(stream, xl, src, dst, as2, ad2, E, N, als, ald, mseg, sseg, ealpha, hB);
    hipMemsetAsync(bnsum, 0, DDIM * sizeof(float), stream);
    hipMemsetAsync(bnsq,  0, DDIM * sizeof(float), stream);
    k_bn_stats<<<512, 256, 0, stream>>>(hB, bnsum, bnsq, N);
    k_bn_apply<<<cdiv64((long long)N * DDIM, 256), 256, 0, stream>>>(
        hB, bnsum, bnsq, bn2g, bn2b, /*residual=*/hA, N);  // hB = leaky(BN(h)) + h_prev

    // ---------------- output projection ----------------
    {
        dim3 grid(OUTD / 64, N / 32);
        k_gemm_f32<<<grid, 256, 0, stream>>>(hB, outW, (float*)d_out, outb, N, OUTD, DDIM);
    }
}